// Net_imp_45148696215617
// MI455X (gfx1250) — compile-verified
//
#include <hip/hip_runtime.h>
#include <hip/hip_bf16.h>
#include <math.h>

#define FIN   128
#define HEADS 8
#define OUT1  8
#define HID   64   // HEADS*OUT1
#define NCLS  40
#define CP    48   // padded class dim (3 WMMA tiles)
#define NEG   0.2f

#define WS1   (FIN + 2)  // padded K-stride for transposed W1 in LDS (bank spread, even)
#define WS2   (HID + 2)  // padded K-stride for transposed W2 in LDS

typedef float v2f __attribute__((ext_vector_type(2)));
typedef float v8f __attribute__((ext_vector_type(8)));

__device__ __forceinline__ float lrelu(float x) { return x > 0.f ? x : NEG * x; }

// order-preserving float -> uint key for atomicMax-based segment max
__device__ __forceinline__ unsigned fkey(float f) {
    unsigned u = __float_as_uint(f);
    return (u & 0x80000000u) ? ~u : (u | 0x80000000u);
}
__device__ __forceinline__ float funkey(unsigned k) {
    return __uint_as_float((k & 0x80000000u) ? (k & 0x7fffffffu) : ~k);
}

// ---------------- GEMM1: h1[N,64] = x[N,128] @ W1[128,64] (WMMA f32 16x16x4) ----------------
// W1 is staged TRANSPOSED in LDS: sW[c*WS1 + k] so each B fragment (K,K+1 pair for one
// column) is one contiguous 8-byte ds_load_b64 straight into an even VGPR pair.
__global__ __launch_bounds__(256) void gemm1_kernel(const float* __restrict__ x,
                                                    const float* __restrict__ W,
                                                    float* __restrict__ h1, int n) {
    __shared__ float sW[HID * WS1];  // 64 cols x 130 = 33.3 KB
    for (int i = threadIdx.x; i < FIN * HID; i += 256) {
        int k = i / HID, c = i % HID;   // coalesced read of W[k][c]
        sW[c * WS1 + k] = W[i];
    }
    __syncthreads();

    const int wave = threadIdx.x >> 5;
    const int lane = threadIdx.x & 31;
    const int r  = lane & 15;
    const int hh = lane >> 4;

    const int row0 = blockIdx.x * 128 + wave * 16;
    const int m  = row0 + r;
    const int ml = m < n ? m : n - 1;  // branchless clamp: EXEC stays all-1 for WMMA
    const float* xr = x + (size_t)ml * FIN;
    const float* wc0 = &sW[(size_t)(0 * 16 + r) * WS1 + 2 * hh];
    const float* wc1 = &sW[(size_t)(1 * 16 + r) * WS1 + 2 * hh];
    const float* wc2 = &sW[(size_t)(2 * 16 + r) * WS1 + 2 * hh];
    const float* wc3 = &sW[(size_t)(3 * 16 + r) * WS1 + 2 * hh];

    v8f c0 = {}, c1 = {}, c2 = {}, c3 = {};
#pragma unroll 8
    for (int k = 0; k < FIN; k += 4) {
        v2f a  = *(const v2f*)&xr[k + 2 * hh];     // global_load_b64
        v2f b0 = *(const v2f*)&wc0[k];             // ds_load_b64, no repack
        v2f b1 = *(const v2f*)&wc1[k];
        v2f b2 = *(const v2f*)&wc2[k];
        v2f b3 = *(const v2f*)&wc3[k];
        c0 = __builtin_amdgcn_wmma_f32_16x16x4_f32(false, a, false, b0, (short)0, c0, false, false);
        c1 = __builtin_amdgcn_wmma_f32_16x16x4_f32(false, a, false, b1, (short)0, c1, false, false);
        c2 = __builtin_amdgcn_wmma_f32_16x16x4_f32(false, a, false, b2, (short)0, c2, false, false);
        c3 = __builtin_amdgcn_wmma_f32_16x16x4_f32(false, a, false, b3, (short)0, c3, false, false);
    }
#pragma unroll
    for (int v = 0; v < 8; ++v) {
        int row = row0 + v + 8 * hh;
        if (row < n) {
            float* o = h1 + (size_t)row * HID;
            o[r]      = c0[v];
            o[16 + r] = c1[v];
            o[32 + r] = c2[v];
            o[48 + r] = c3[v];
        }
    }
}

// ---------------- GEMM2: h2[N,48] = hmid[N,64] @ W2pad[64,48] (WMMA f32 16x16x4) ----------------
__global__ __launch_bounds__(256) void gemm2_kernel(const float* __restrict__ hm,
                                                    const float* __restrict__ W2,
                                                    float* __restrict__ h2, int n) {
    __shared__ float sW[CP * WS2];  // 48 cols x 66 = 12.4 KB, transposed + zero-padded
    for (int i = threadIdx.x; i < HID * CP; i += 256) {
        int k = i / CP, c = i % CP;
        sW[c * WS2 + k] = (c < NCLS) ? W2[k * NCLS + c] : 0.f;
    }
    __syncthreads();

    const int wave = threadIdx.x >> 5;
    const int lane = threadIdx.x & 31;
    const int r  = lane & 15;
    const int hh = lane >> 4;

    const int row0 = blockIdx.x * 128 + wave * 16;
    const int m  = row0 + r;
    const int ml = m < n ? m : n - 1;
    const float* xr = hm + (size_t)ml * HID;
    const float* wc0 = &sW[(size_t)(0 * 16 + r) * WS2 + 2 * hh];
    const float* wc1 = &sW[(size_t)(1 * 16 + r) * WS2 + 2 * hh];
    const float* wc2 = &sW[(size_t)(2 * 16 + r) * WS2 + 2 * hh];

    v8f c0 = {}, c1 = {}, c2 = {};
#pragma unroll 8
    for (int k = 0; k < HID; k += 4) {
        v2f a  = *(const v2f*)&xr[k + 2 * hh];
        v2f b0 = *(const v2f*)&wc0[k];
        v2f b1 = *(const v2f*)&wc1[k];
        v2f b2 = *(const v2f*)&wc2[k];
        c0 = __builtin_amdgcn_wmma_f32_16x16x4_f32(false, a, false, b0, (short)0, c0, false, false);
        c1 = __builtin_amdgcn_wmma_f32_16x16x4_f32(false, a, false, b1, (short)0, c1, false, false);
        c2 = __builtin_amdgcn_wmma_f32_16x16x4_f32(false, a, false, b2, (short)0, c2, false, false);
    }
#pragma unroll
    for (int v = 0; v < 8; ++v) {
        int row = row0 + v + 8 * hh;
        if (row < n) {
            float* o = h2 + (size_t)row * CP;
            o[r]      = c0[v];
            o[16 + r] = c1[v];
            o[32 + r] = c2[v];
        }
    }
}

// ---------------- per-node attention coefficients, layer 1 ----------------
__global__ void alpha1_kernel(const float* __restrict__ h1,
                              const float* __restrict__ attS, const float* __restrict__ attD,
                              float* __restrict__ as1, float* __restrict__ ad1, int n) {
    int t = blockIdx.x * blockDim.x + threadIdx.x;
    if (t >= n * HEADS) return;
    int node = t >> 3, h = t & 7;
    const float* hr = h1 + (size_t)node * HID + h * OUT1;
    const float* sa = attS + h * OUT1;
    const float* da = attD + h * OUT1;
    float s = 0.f, d = 0.f;
#pragma unroll
    for (int o = 0; o < OUT1; ++o) { s += hr[o] * sa[o]; d += hr[o] * da[o]; }
    as1[t] = s;
    ad1[t] = d;
}

// ---------------- layer-1 edge passes ----------------
__global__ void l1_edge_max(const int* __restrict__ ei, int E, int Etot,
                            const float* __restrict__ as1, const float* __restrict__ ad1,
                            unsigned* __restrict__ m1) {
    int idx = blockIdx.x * blockDim.x + threadIdx.x;
    int edge = idx >> 3, h = idx & 7;
    if (edge >= Etot) return;
    int s, d;
    if (edge < E) { s = ei[edge]; d = ei[E + edge]; } else { s = d = edge - E; }
    float e = lrelu(as1[s * HEADS + h] + ad1[d * HEADS + h]);
    atomicMax(&m1[d * HEADS + h], fkey(e));
}

__global__ void l1_edge_acc(const int* __restrict__ ei, int E, int Etot,
                            const float* __restrict__ h1,
                            const float* __restrict__ as1, const float* __restrict__ ad1,
                            const unsigned* __restrict__ m1,
                            float* __restrict__ den1, float* __restrict__ acc1) {
    int idx = blockIdx.x * blockDim.x + threadIdx.x;
    int edge = idx >> 3, h = idx & 7;
    if (edge >= Etot) return;
    int s, d;
    if (edge < E) { s = ei[edge]; d = ei[E + edge]; } else { s = d = edge - E; }
    float e  = lrelu(as1[s * HEADS + h] + ad1[d * HEADS + h]);
    float ex = __expf(e - funkey(m1[d * HEADS + h]));
    atomicAdd(&den1[d * HEADS + h], ex);
    const float* hs = h1 + (size_t)s * HID + h * OUT1;
    float* ac = acc1 + (size_t)d * HID + h * OUT1;
#pragma unroll
    for (int o = 0; o < OUT1; ++o) atomicAdd(&ac[o], ex * hs[o]);
}

// ---------------- layer-1 finish: hmid = lrelu(acc/den + b1), in place over h1 ----------------
__global__ void l1_finish(float* __restrict__ h1io,
                          const float* __restrict__ acc1, const float* __restrict__ den1,
                          const float* __restrict__ b1, int n) {
    int t = blockIdx.x * blockDim.x + threadIdx.x;
    if (t >= n * HID) return;
    int node = t >> 6, j = t & 63;
    float v = acc1[t] / den1[node * HEADS + (j >> 3)] + b1[j];
    h1io[t] = lrelu(v);
}

// ---------------- per-node attention coefficients, layer 2 ----------------
__global__ void alpha2_kernel(const float* __restrict__ h2,
                              const float* __restrict__ attS, const float* __restrict__ attD,
                              float* __restrict__ as2, float* __restrict__ ad2, int n) {
    int t = blockIdx.x * blockDim.x + threadIdx.x;
    if (t >= n) return;
    const float* hr = h2 + (size_t)t * CP;
    float s = 0.f, d = 0.f;
#pragma unroll
    for (int c = 0; c < NCLS; ++c) { s += hr[c] * attS[c]; d += hr[c] * attD[c]; }
    as2[t] = s;
    ad2[t] = d;
}

// ---------------- layer-2 edge passes ----------------
__global__ void l2_edge_max(const int* __restrict__ ei, int E, int Etot,
                            const float* __restrict__ as2, const float* __restrict__ ad2,
                            unsigned* __restrict__ m2) {
    int edge = blockIdx.x * blockDim.x + threadIdx.x;
    if (edge >= Etot) return;
    int s, d;
    if (edge < E) { s = ei[edge]; d = ei[E + edge]; } else { s = d = edge - E; }
    float e = lrelu(as2[s] + ad2[d]);
    atomicMax(&m2[d], fkey(e));
}

__global__ void l2_edge_acc(const int* __restrict__ ei, int E, int Etot,
                            const float* __restrict__ h2,
                            const float* __restrict__ as2, const float* __restrict__ ad2,
                            const unsigned* __restrict__ m2,
                            float* __restrict__ den2, float* __restrict__ acc2) {
    int idx = blockIdx.x * blockDim.x + threadIdx.x;
    int edge = idx >> 3, sub = idx & 7;  // 8 threads/edge, 5 classes each
    if (edge >= Etot) return;
    int s, d;
    if (edge < E) { s = ei[edge]; d = ei[E + edge]; } else { s = d = edge - E; }
    float e  = lrelu(as2[s] + ad2[d]);
    float ex = __expf(e - funkey(m2[d]));
    if (sub == 0) atomicAdd(&den2[d], ex);
    const float* hs = h2 + (size_t)s * CP + sub * 5;
    float* ac = acc2 + (size_t)d * NCLS + sub * 5;
#pragma unroll
    for (int j = 0; j < 5; ++j) atomicAdd(&ac[j], ex * hs[j]);
}

// ---------------- final: out = log_softmax(acc/den + b2) ----------------
__global__ void final_kernel(const float* __restrict__ acc2, const float* __restrict__ den2,
                             const float* __restrict__ b2, float* __restrict__ out, int n) {
    int t = blockIdx.x * blockDim.x + threadIdx.x;
    if (t >= n) return;
    const float* ac = acc2 + (size_t)t * NCLS;
    float dn = den2[t];
    float o[NCLS];
    float mx = -3.4e38f;
#pragma unroll
    for (int c = 0; c < NCLS; ++c) {
        o[c] = ac[c] / dn + b2[c];
        mx = fmaxf(mx, o[c]);
    }
    float sum = 0.f;
#pragma unroll
    for (int c = 0; c < NCLS; ++c) sum += __expf(o[c] - mx);
    float l = __logf(sum) + mx;
    float* op = out + (size_t)t * NCLS;
#pragma unroll
    for (int c = 0; c < NCLS; ++c) op[c] = o[c] - l;
}

extern "C" void kernel_launch(void* const* d_in, const int* in_sizes, int n_in,
                              void* d_out, int out_size, void* d_ws, size_t ws_size,
                              hipStream_t stream) {
    const float* x     = (const float*)d_in[0];
    const int*   ei    = (const int*)  d_in[1];
    const float* W1    = (const float*)d_in[2];
    const float* attS1 = (const float*)d_in[3];
    const float* attD1 = (const float*)d_in[4];
    const float* b1    = (const float*)d_in[5];
    const float* W2    = (const float*)d_in[6];
    const float* attS2 = (const float*)d_in[7];
    const float* attD2 = (const float*)d_in[8];
    const float* b2    = (const float*)d_in[9];
    float* out = (float*)d_out;

    const int n    = in_sizes[0] / FIN;   // 100000
    const int E    = in_sizes[1] / 2;     // 1600000
    const int Etot = E + n;               // with self-loops

    // workspace layout (floats); zeroed region is contiguous at the end
    float* w = (float*)d_ws;
    float* h1  = w; w += (size_t)n * HID;
    float* as1 = w; w += (size_t)n * HEADS;
    float* ad1 = w; w += (size_t)n * HEADS;
    float* h2  = w; w += (size_t)n * CP;
    float* as2 = w; w += n;
    float* ad2 = w; w += n;
    float* zbase = w;
    unsigned* m1 = (unsigned*)w; w += (size_t)n * HEADS;
    float* den1  = w;            w += (size_t)n * HEADS;
    float* acc1  = w;            w += (size_t)n * HID;
    unsigned* m2 = (unsigned*)w; w += n;
    float* den2  = w;            w += n;
    float* acc2  = w;            w += (size_t)n * NCLS;
    size_t zbytes = (size_t)((char*)w - (char*)zbase);
    hipMemsetAsync(zbase, 0, zbytes, stream);  // m=-inf (key 0), den=0, acc=0

    const int gb = (n + 127) / 128;  // 128 rows per block (8 waves x 16)
    gemm1_kernel<<<gb, 256, 0, stream>>>(x, W1, h1, n);
    alpha1_kernel<<<(n * HEADS + 255) / 256, 256, 0, stream>>>(h1, attS1, attD1, as1, ad1, n);
    l1_edge_max<<<(Etot * HEADS + 255) / 256, 256, 0, stream>>>(ei, E, Etot, as1, ad1, m1);
    l1_edge_acc<<<(Etot * HEADS + 255) / 256, 256, 0, stream>>>(ei, E, Etot, h1, as1, ad1, m1, den1, acc1);
    l1_finish<<<(n * HID + 255) / 256, 256, 0, stream>>>(h1, acc1, den1, b1, n);
    gemm2_kernel<<<gb, 256, 0, stream>>>(h1, W2, h2, n);
    alpha2_kernel<<<(n + 255) / 256, 256, 0, stream>>>(h2, attS2, attD2, as2, ad2, n);
    l2_edge_max<<<(Etot + 255) / 256, 256, 0, stream>>>(ei, E, Etot, as2, ad2, m2);
    l2_edge_acc<<<(Etot * 8 + 255) / 256, 256, 0, stream>>>(ei, E, Etot, h2, as2, ad2, m2, den2, acc2);
    final_kernel<<<(n + 255) / 256, 256, 0, stream>>>(acc2, den2, b2, out, n);
}